// IndexLinear_25125558682018
// MI455X (gfx1250) — compile-verified
//
#include <hip/hip_runtime.h>

typedef __attribute__((ext_vector_type(2))) float v2f;
typedef __attribute__((ext_vector_type(4))) float v4f;
typedef __attribute__((ext_vector_type(8))) float v8f;

#define N_TOK   8192
#define IN_DIM  1024
#define OUT_DIM 1024
#define N_EXP   16
#define MAX_TILES 544   // >= 8192/16 + 16 padding tiles

// d_ws int layout
#define WS_COUNTS   0               // [16]
#define WS_CURSORS  16              // [16]
#define WS_PADBASE  32              // [16] tile-base per expert
#define WS_NUMTILES 48              // [1]
#define WS_TILEEXP  64              // [MAX_TILES]
#define WS_PERM     (64 + MAX_TILES) // [MAX_TILES*16] padded grouped token ids (-1 = pad)

__global__ __launch_bounds__(256) void k_init(int* __restrict__ ws) {
    int i = blockIdx.x * 256 + threadIdx.x;
    if (i < MAX_TILES * 16) ws[WS_PERM + i] = -1;
    if (i < 32) ws[WS_COUNTS + i] = 0;   // counts + cursors contiguous
}

__global__ __launch_bounds__(256) void k_hist(const int* __restrict__ ind,
                                              int* __restrict__ ws) {
    int n = blockIdx.x * 256 + threadIdx.x;
    if (n < N_TOK) {
        int e = ind[n] & (N_EXP - 1);
        atomicAdd(&ws[WS_COUNTS + e], 1);
    }
}

__global__ void k_plan(int* __restrict__ ws) {
    if (threadIdx.x == 0 && blockIdx.x == 0) {
        int tb = 0;
        for (int e = 0; e < N_EXP; ++e) {
            ws[WS_PADBASE + e] = tb;
            int nt = (ws[WS_COUNTS + e] + 15) >> 4;
            for (int i = 0; i < nt; ++i) ws[WS_TILEEXP + tb + i] = e;
            tb += nt;
        }
        ws[WS_NUMTILES] = tb;
    }
}

__global__ __launch_bounds__(256) void k_scatter(const int* __restrict__ ind,
                                                 int* __restrict__ ws) {
    int n = blockIdx.x * 256 + threadIdx.x;
    if (n < N_TOK) {
        int e = ind[n] & (N_EXP - 1);
        int pos = atomicAdd(&ws[WS_CURSORS + e], 1);
        ws[WS_PERM + ws[WS_PADBASE + e] * 16 + pos] = n;
    }
}

// One wave = 16 tokens x 64 output columns, fp32 WMMA 16x16x4, K=1024.
__global__ __launch_bounds__(128) void k_gemm(const float* __restrict__ x,
                                              const float* __restrict__ W,
                                              const float* __restrict__ bias,
                                              float* __restrict__ out,
                                              const int* __restrict__ ws) {
    const int t = blockIdx.x;
    if (t >= ws[WS_NUMTILES]) return;          // wave-uniform: EXEC stays full
    const int e    = ws[WS_TILEEXP + t];
    const int lane = threadIdx.x & 31;
    const int wave = threadIdx.x >> 5;
    const int m    = lane & 15;                // token row for A / out column for B,D
    const int h    = lane >> 4;                // K half: carries K offsets {4h..4h+3}
    const int* __restrict__ perm = ws + WS_PERM + t * 16;

    const int tok  = perm[m];
    const int tokS = (tok >= 0) ? tok : 0;     // clamp pad rows for loads

    const float* aPtr = x + (size_t)tokS * IN_DIM + 4 * h;
    const int nBase = (blockIdx.y * 4 + wave) * 64;
    const float* wExp = W + (size_t)e * ((size_t)OUT_DIM * IN_DIM);

    const float* wPtr[4];
    float bv[4];
    #pragma unroll
    for (int nt = 0; nt < 4; ++nt) {
        int col = nBase + nt * 16 + m;
        wPtr[nt] = wExp + (size_t)col * IN_DIM + 4 * h;
        bv[nt] = bias[e * OUT_DIM + col];
    }

    v8f acc[4];
    const v8f zero = {0.f, 0.f, 0.f, 0.f, 0.f, 0.f, 0.f, 0.f};
    #pragma unroll
    for (int nt = 0; nt < 4; ++nt) acc[nt] = zero;

    for (int k = 0; k < IN_DIM; k += 8) {
        v4f av = *(const v4f*)(aPtr + k);
        v2f a01 = {av.x, av.y};
        v2f a23 = {av.z, av.w};
        #pragma unroll
        for (int nt = 0; nt < 4; ++nt) {
            v4f wv = *(const v4f*)(wPtr[nt] + k);
            v2f w01 = {wv.x, wv.y};
            v2f w23 = {wv.z, wv.w};
            acc[nt] = __builtin_amdgcn_wmma_f32_16x16x4_f32(
                false, a01, false, w01, (short)0, acc[nt], false, false);
            acc[nt] = __builtin_amdgcn_wmma_f32_16x16x4_f32(
                false, a23, false, w23, (short)0, acc[nt], false, false);
        }
    }

    // D layout: lane column N = m; VGPR r holds row M = 8*h + r.
    const int row0 = 8 * h;
    #pragma unroll
    for (int r = 0; r < 8; ++r) {
        int trow = perm[row0 + r];
        if (trow >= 0) {
            float* orow = out + (size_t)trow * OUT_DIM + nBase + m;
            orow[0]  = acc[0][r] + bv[0];
            orow[16] = acc[1][r] + bv[1];
            orow[32] = acc[2][r] + bv[2];
            orow[48] = acc[3][r] + bv[3];
        }
    }
}

extern "C" void kernel_launch(void* const* d_in, const int* in_sizes, int n_in,
                              void* d_out, int out_size, void* d_ws, size_t ws_size,
                              hipStream_t stream) {
    const float* x   = (const float*)d_in[0];
    const int*   ind = (const int*)d_in[1];
    const float* W   = (const float*)d_in[2];
    const float* b   = (const float*)d_in[3];
    float* out = (float*)d_out;
    int*   ws  = (int*)d_ws;

    k_init<<<(MAX_TILES * 16 + 255) / 256, 256, 0, stream>>>(ws);
    k_hist<<<N_TOK / 256, 256, 0, stream>>>(ind, ws);
    k_plan<<<1, 32, 0, stream>>>(ws);
    k_scatter<<<N_TOK / 256, 256, 0, stream>>>(ind, ws);
    k_gemm<<<dim3(MAX_TILES, 4, 1), dim3(128, 1, 1), 0, stream>>>(x, W, b, out, ws);
}